// Decoding_model_1932735283325
// MI455X (gfx1250) — compile-verified
//
#include <hip/hip_runtime.h>
#include <math.h>

// Problem constants (match reference)
#define B_   64
#define N_   1024
#define M_   512
#define DC_  6
#define IT_  5
#define BM_  (B_ * M_)    // 32768 check-rows per iteration
#define BN_  (B_ * N_)    // 65536 variables per iteration
#define E_   (M_ * DC_)   // 3072 edges (structure)
#define CAP_ 64           // per-column adjacency capacity (true max deg ~ 12)

typedef float v2f __attribute__((ext_vector_type(2)));
typedef float v8f __attribute__((ext_vector_type(8)));

// ---------------------------------------------------------------------------
// params[0..4] = w = W1 @ W2  (the model's only real matmul, on the WMMA unit)
// params[5]    = softplus(bit_w1), params[6] = softplus(bit_w2)
// A-layout (32-bit A 16x4): lane L holds row (L&15); K = 2*(L>=16) + comp.
// B-layout (32-bit B 4x16): lane L holds col (L&15); K = v + 2*(L>=16),
// i.e. same {x->K=2h, y->K=2h+1} per-lane packing as A.
// D (16x16 f32 C/D layout): D[i][0] lives in VGPR i of lane 0.
// ---------------------------------------------------------------------------
__global__ __launch_bounds__(32) void compute_params_kernel(
    const float* __restrict__ W1,   // [5,4] row-major
    const float* __restrict__ W2,   // [4,1]
    const float* __restrict__ bw1,
    const float* __restrict__ bw2,
    float* __restrict__ params)
{
  int lane = threadIdx.x;
  int row  = lane & 15;
  int kb   = (lane >> 4) * 2;   // K base for this lane half

  v2f a, b;
  a.x = (row < 5) ? W1[row * 4 + kb]     : 0.0f;
  a.y = (row < 5) ? W1[row * 4 + kb + 1] : 0.0f;
  int col = lane & 15;
  b.x = (col == 0) ? W2[kb]     : 0.0f;
  b.y = (col == 0) ? W2[kb + 1] : 0.0f;

  v8f c = {0.f, 0.f, 0.f, 0.f, 0.f, 0.f, 0.f, 0.f};
  // (neg_a, A, neg_b, B, c_mod, C, reuse_a, reuse_b)
  c = __builtin_amdgcn_wmma_f32_16x16x4_f32(false, a, false, b,
                                            (short)0, c, false, false);
  if (lane == 0) {
    params[0] = c[0];  // w_0 = D[0][0]
    params[1] = c[1];
    params[2] = c[2];
    params[3] = c[3];
    params[4] = c[4];
    params[5] = log1pf(expf(bw1[0]));  // softplus
    params[6] = log1pf(expf(bw2[0]));
  }
}

// ---------------------------------------------------------------------------
// Deterministic CSC build: one thread per variable column scans the (tiny,
// uniform-broadcast) edge list. csc entries store the planar cv index base
// (j*BM + m); marginalize adds b*M.
// ---------------------------------------------------------------------------
__global__ __launch_bounds__(256) void build_csc_kernel(
    const int* __restrict__ row_cols,   // [M,6]
    int* __restrict__ deg,              // [N]
    int* __restrict__ csc)              // [N,CAP]
{
  int n = blockIdx.x * blockDim.x + threadIdx.x;
  if (n >= N_) return;
  int d = 0;
  for (int e = 0; e < E_; ++e) {        // uniform address -> broadcast loads
    if (row_cols[e] == n) {
      int m = e / DC_;
      int j = e - m * DC_;
      if (d < CAP_) csc[n * CAP_ + d] = j * BM_ + m;
      ++d;
    }
  }
  deg[n] = (d < CAP_) ? d : CAP_;
}

// ---------------------------------------------------------------------------
// out[0] = soft_input; colsum = 0 (cv starts at zero)
// ---------------------------------------------------------------------------
__global__ __launch_bounds__(256) void init_kernel(
    const float* __restrict__ soft,
    float* __restrict__ out0,
    float* __restrict__ colsum)
{
  int t = blockIdx.x * blockDim.x + threadIdx.x;
  if (t < BN_) {
    out0[t]   = soft[t];
    colsum[t] = 0.0f;
  }
}

// ---------------------------------------------------------------------------
// Check-node update: one lane = one (b, m) row. Fully unrolled, register-only.
//   vc_j   = (colsum[b,c_j] + sp1*soft[b,c_j]) - cv_j
//   P      = prod_j sign(vc_j)                       (0-propagating, as jnp.sign)
//   rank_j = position of |vc_j| in ascending sort (ties broken by index)
//   vals(rank r) = sum_{i<r} w_i*a_i + sum_{i>=r} w_i*a_{i+1}   (LOO sorted dot)
//   cv_j'  = P * sign(vc_j) * vals(rank_j)
// ---------------------------------------------------------------------------
__global__ __launch_bounds__(256) void check_kernel(
    const float* __restrict__ soft,
    const int*   __restrict__ row_cols,
    const float* __restrict__ params,
    const float* __restrict__ colsum,
    float*       __restrict__ cv,       // planar [6][B*M]
    int first)
{
  int t = blockIdx.x * blockDim.x + threadIdx.x;
  if (t >= BM_) return;
  int b = t >> 9;          // / M_ (512)
  int m = t & (M_ - 1);

  const float* sb = soft   + b * N_;
  const float* cb = colsum + b * N_;

  float w[5];
#pragma unroll
  for (int i = 0; i < 5; ++i) w[i] = params[i];   // uniform -> scalar loads
  float sp1 = params[5];

  float vc[6];
#pragma unroll
  for (int j = 0; j < 6; ++j) {
    int c = row_cols[m * 6 + j];
    float temp = cb[c] + sp1 * sb[c];
    float old  = first ? 0.0f : cv[j * BM_ + t];
    vc[j] = temp - old;
  }

  float sgn[6], nz[6];
  float P = 1.0f;
#pragma unroll
  for (int j = 0; j < 6; ++j) {
    float v = vc[j];
    float s = (v > 0.0f) ? 1.0f : ((v < 0.0f) ? -1.0f : 0.0f);
    sgn[j] = s;
    P *= s;
    nz[j] = fabsf(v);
  }

  // ranks by comparison counting (distinct, tie -> lower index first)
  int r[6];
#pragma unroll
  for (int j = 0; j < 6; ++j) {
    int rj = 0;
#pragma unroll
    for (int k = 0; k < 6; ++k)
      rj += (nz[k] < nz[j]) || ((nz[k] == nz[j]) && (k < j));
    r[j] = rj;
  }

  // a[i] = magnitude of rank i (branch-free select; all indices compile-time)
  float a[6];
#pragma unroll
  for (int i = 0; i < 6; ++i) {
    float ai = 0.0f;
#pragma unroll
    for (int k = 0; k < 6; ++k) ai = (r[k] == i) ? nz[k] : ai;
    a[i] = ai;
  }

  // prefix/suffix dot with w over the sorted array, skipping rank r
  float hi[6];
  hi[5] = 0.0f;
#pragma unroll
  for (int rr = 4; rr >= 0; --rr) hi[rr] = hi[rr + 1] + w[rr] * a[rr + 1];
  float vr[6];
  vr[0] = hi[0];
  float lo = 0.0f;
#pragma unroll
  for (int rr = 1; rr < 6; ++rr) { lo += w[rr - 1] * a[rr - 1]; vr[rr] = lo + hi[rr]; }

#pragma unroll
  for (int j = 0; j < 6; ++j) {
    int rj = r[j];
    float v = vr[0];
    v = (rj == 1) ? vr[1] : v;
    v = (rj == 2) ? vr[2] : v;
    v = (rj == 3) ? vr[3] : v;
    v = (rj == 4) ? vr[4] : v;
    v = (rj == 5) ? vr[5] : v;
    cv[j * BM_ + t] = P * sgn[j] * v;
  }
}

// ---------------------------------------------------------------------------
// Marginalize + maintain colsum: deterministic fixed-order gather (no float
// atomics).  colsum[b,n] = sum_{edges on n} cv';  out = colsum + sp2*soft.
// ---------------------------------------------------------------------------
__global__ __launch_bounds__(256) void marg_kernel(
    const float* __restrict__ soft,
    const float* __restrict__ cv,
    const int*   __restrict__ deg,
    const int*   __restrict__ csc,
    const float* __restrict__ params,
    float*       __restrict__ colsum,
    float*       __restrict__ out_slice)
{
  int t = blockIdx.x * blockDim.x + threadIdx.x;
  if (t >= BN_) return;
  int b = t >> 10;           // / N_ (1024)
  int n = t & (N_ - 1);

  float sp2  = params[6];
  int   d    = deg[n];
  int   base = b * M_;
  float s = 0.0f;
  for (int i = 0; i < d; ++i) {
    int idx = csc[n * CAP_ + i];      // j*BM + m
    s += cv[idx + base];
  }
  colsum[t]    = s;
  out_slice[t] = s + sp2 * soft[t];
}

// ---------------------------------------------------------------------------
extern "C" void kernel_launch(void* const* d_in, const int* in_sizes, int n_in,
                              void* d_out, int out_size, void* d_ws, size_t ws_size,
                              hipStream_t stream)
{
  const float* soft     = (const float*)d_in[0];
  // d_in[1] = labels (unused), d_in[2] = H (dense copy, unused: row_cols is H)
  const int*   row_cols = (const int*)d_in[3];
  const float* W1       = (const float*)d_in[4];
  const float* W2       = (const float*)d_in[5];
  const float* bw1      = (const float*)d_in[6];
  const float* bw2      = (const float*)d_in[7];
  float*       out      = (float*)d_out;       // [IT_+1, B, N]

  // Workspace layout (all initialized every call; ~1.31 MB)
  float* cv     = (float*)d_ws;                // 6*BM floats
  float* colsum = cv + 6 * BM_;                // BN floats
  float* params = colsum + BN_;                // 8 floats
  int*   deg    = (int*)(params + 8);          // N ints
  int*   csc    = deg + N_;                    // N*CAP ints

  compute_params_kernel<<<1, 32, 0, stream>>>(W1, W2, bw1, bw2, params);
  build_csc_kernel<<<N_ / 256, 256, 0, stream>>>(row_cols, deg, csc);
  init_kernel<<<BN_ / 256, 256, 0, stream>>>(soft, out, colsum);

  for (int it = 0; it < IT_; ++it) {
    check_kernel<<<BM_ / 256, 256, 0, stream>>>(
        soft, row_cols, params, colsum, cv, (it == 0) ? 1 : 0);
    marg_kernel<<<BN_ / 256, 256, 0, stream>>>(
        soft, cv, deg, csc, params, colsum, out + (size_t)(it + 1) * BN_);
  }
}